// CapsuleLayer_28544352649849
// MI455X (gfx1250) — compile-verified
//
#include <hip/hip_runtime.h>

// ---------------------------------------------------------------------------
// CapsuleLayer on MI455X (gfx1250, wave32).
// Exact algebraic collapse of the routing loop (see analysis):
//   out[b,i,k] = squash_k( (1/256) * sum_m inputs[b,m] * Wsum[m,k] ),
//   Wsum[m,k]  = sum_i W[i,m,k]
// Bandwidth-bound: ~67MB read + ~134MB write => ~8.6us @ 23.3 TB/s.
// GEMM step uses V_WMMA_F32_16X16X4_F32 (fp32-exact tensor path).
// ---------------------------------------------------------------------------

typedef __attribute__((ext_vector_type(2))) float v2f;
typedef __attribute__((ext_vector_type(4))) float v4f;
typedef __attribute__((ext_vector_type(8))) float v8f;

#define NK 256          // num_capsule == dim_capsule == 256
#define BATCH 512

// ---- Kernel 1: partial reduce of W[i,m,k] over i, in 8 chunks of 32 -------
// 131072 threads, each owns one (chunk, float4-of-(m,k)) and sums 32 planes.
__global__ __launch_bounds__(256)
void wsum_partial_kernel(const float* __restrict__ W, float* __restrict__ partials) {
    int gid = blockIdx.x * 256 + threadIdx.x;   // 0 .. 131071
    int c   = gid >> 14;                        // chunk 0..7
    int p   = gid & 16383;                      // float4 index in 64K plane
    const v4f* Wv = (const v4f*)W;
    v4f s = {};
    int base = (c * 32) * 16384 + p;
#pragma unroll 8
    for (int ii = 0; ii < 32; ++ii) {
        v4f w = Wv[base + ii * 16384];
        s.x += w.x; s.y += w.y; s.z += w.z; s.w += w.w;
    }
    ((v4f*)partials)[c * 16384 + p] = s;
}

// ---- Kernel 2: combine the 8 partial planes -> Wsum[256,256] --------------
__global__ __launch_bounds__(256)
void wsum_final_kernel(const float* __restrict__ partials, float* __restrict__ Wsum) {
    int p = blockIdx.x * 256 + threadIdx.x;     // 0 .. 16383
    const v4f* Pv = (const v4f*)partials;
    v4f s = Pv[p];
#pragma unroll
    for (int c = 1; c < 8; ++c) {
        v4f w = Pv[c * 16384 + p];
        s.x += w.x; s.y += w.y; s.z += w.z; s.w += w.w;
    }
    ((v4f*)Wsum)[p] = s;
}

// ---- Kernel 3: U = (inputs @ Wsum) / 256 via V_WMMA_F32_16X16X4_F32 -------
// One wave per 16x16 output tile. A (16x4 f32): lanes 0-15 hold K=k,k+1;
// lanes 16-31 hold K=k+2,k+3 (ISA 7.12.2). B (4x16): mirrored over N.
// C/D: VGPR r -> row r (lanes 0-15) / row r+8 (lanes 16-31).
__global__ __launch_bounds__(32)
void gemm_wmma_kernel(const float* __restrict__ A,    // inputs [512,256]
                      const float* __restrict__ Bm,   // Wsum   [256,256]
                      float* __restrict__ U) {        // u      [512,256]
    int lane = threadIdx.x;
    int half = lane >> 4;          // 0: K pair {0,1}, 1: K pair {2,3}
    int lm   = lane & 15;
    int rowA = blockIdx.x * 16 + lm;   // M (batch) index for A
    int colB = blockIdx.y * 16 + lm;   // N (k) index for B
    const float* aBase = A  + rowA * NK + 2 * half;
    const float* bBase = Bm + 2 * half * NK + colB;
    v8f acc = {};
#pragma unroll 4
    for (int k = 0; k < NK; k += 4) {
        v2f a = *(const v2f*)(aBase + k);            // K = k+2h, k+1+2h
        v2f b;
        b.x = bBase[k * NK];                         // B[k+2h][colB]
        b.y = bBase[k * NK + NK];                    // B[k+1+2h][colB]
        acc = __builtin_amdgcn_wmma_f32_16x16x4_f32(
            /*neg_a=*/false, a, /*neg_b=*/false, b,
            /*c_mod=*/(short)0, acc, /*reuse_a=*/false, /*reuse_b=*/false);
    }
    const float scale = 1.0f / 256.0f;
    float* uBase = U + (blockIdx.x * 16 + 8 * half) * NK + blockIdx.y * 16 + lm;
#pragma unroll
    for (int r = 0; r < 8; ++r)
        uBase[r * NK] = acc[r] * scale;
}

// ---- Kernel 4: squash each row of U -> V ----------------------------------
__global__ __launch_bounds__(256)
void squash_kernel(const float* __restrict__ U, float* __restrict__ V) {
    __shared__ float red[256];
    int b = blockIdx.x;
    int t = threadIdx.x;
    float x = U[b * NK + t];
    red[t] = x * x;
    __syncthreads();
#pragma unroll
    for (int off = 128; off > 0; off >>= 1) {
        if (t < off) red[t] += red[t + off];
        __syncthreads();
    }
    float s2 = red[0];
    float sc = (s2 / (1.0f + s2)) * rsqrtf(s2 + 1e-7f);
    V[b * NK + t] = sc * x;
}

// ---- Kernel 5: broadcast V[b,:] over the i axis -> out[b,i,k] -------------
// 134MB streaming write; reads hit L0/L2 (V is 512KB). Nontemporal stores.
__global__ __launch_bounds__(256)
void broadcast_kernel(const float* __restrict__ V, float* __restrict__ out) {
    int idx = blockIdx.x * 256 + threadIdx.x;   // float4 index, 0 .. 8388607
    int k4  = idx & 63;                         // 64 float4 per row
    int b   = idx >> 14;                        // (idx >> 6) >> 8
    v4f val = ((const v4f*)V)[b * 64 + k4];
    __builtin_nontemporal_store(val, ((v4f*)out) + idx);
}

extern "C" void kernel_launch(void* const* d_in, const int* in_sizes, int n_in,
                              void* d_out, int out_size, void* d_ws, size_t ws_size,
                              hipStream_t stream) {
    const float* inputs = (const float*)d_in[0];   // [512, 256]
    const float* W      = (const float*)d_in[1];   // [256, 256, 256]
    float* out = (float*)d_out;                    // [512, 256, 256]
    float* ws  = (float*)d_ws;

    // workspace layout (floats): partials 524288 | Wsum 65536 | U 131072 | V 131072
    float* partials = ws;
    float* Wsum     = ws + 524288;
    float* U        = Wsum + 65536;
    float* V        = U + 131072;

    wsum_partial_kernel<<<512, 256, 0, stream>>>(W, partials);
    wsum_final_kernel  <<<64,  256, 0, stream>>>(partials, Wsum);
    dim3 gGemm(BATCH / 16, NK / 16);               // (32, 16) tiles, 1 wave each
    gemm_wmma_kernel   <<<gGemm, 32, 0, stream>>>(inputs, Wsum, U);
    squash_kernel      <<<BATCH, 256, 0, stream>>>(U, V);
    broadcast_kernel   <<<32768, 256, 0, stream>>>(V, out);
}